// DTCWTForward_42709154791432
// MI455X (gfx1250) — compile-verified
//
#include <hip/hip_runtime.h>

typedef float v2f __attribute__((ext_vector_type(2)));
typedef float v8f __attribute__((ext_vector_type(8)));

#define SQ2 0.70710678118654752440f

// --- gfx1250 async memory->LDS path (guarded: host pass / older toolchains fall back) ---
#if defined(__HIP_DEVICE_COMPILE__) && defined(__gfx1250__) && \
    __has_builtin(__builtin_amdgcn_global_load_async_to_lds_b32) && \
    __has_builtin(__builtin_amdgcn_s_wait_asynccnt)
#define HAVE_ASYNC_LDS 1
#else
#define HAVE_ASYNC_LDS 0
#endif

__device__ __forceinline__ void async_cp_b32(const float* g, float* l) {
#if HAVE_ASYNC_LDS
    __builtin_amdgcn_global_load_async_to_lds_b32(
        (__attribute__((address_space(1))) int*)const_cast<float*>(g),
        (__attribute__((address_space(3))) int*)l,
        0, 0);
#else
    *l = *g;
#endif
}

__device__ __forceinline__ void async_join() {
#if HAVE_ASYNC_LDS
    __builtin_amdgcn_s_wait_asynccnt(0);
#endif
}

__device__ __forceinline__ int refl(int i, int L) {
    // numpy 'symmetric' reflection (edge included); pads << L so one fold suffices
    if (i < 0) i = -i - 1;
    if (i >= L) i = 2 * L - 1 - i;
    return i;
}

// select h[d] from a register array with an unrolled cndmask chain (prep kernel only)
template <int M>
__device__ __forceinline__ float tapsel(const float* h, int d) {
    float v = 0.f;
#pragma unroll
    for (int t = 0; t < M; ++t) v = (d == t) ? h[t] : v;
    return v;
}

// ---------------------------------------------------------------------------
// Prep: build the tile-invariant Toeplitz A-fragments for the level-1 column
// filters, in WMMA A-layout (lane -> row=lane&15, K-half=(lane>>4)*2), one
// float4 {a0.x,a0.y,a1.x,a1.y} per (chunk,lane). 6 chunks x 32 lanes.
// ---------------------------------------------------------------------------
__global__ __launch_bounds__(32)
void dtcwt_prep(const float* __restrict__ h0o, const float* __restrict__ h1o,
                float* __restrict__ tA)
{
    const int lane = threadIdx.x;
    float f0[5], f1[7];
#pragma unroll
    for (int t = 0; t < 5; ++t) f0[t] = h0o[4 - t];
#pragma unroll
    for (int t = 0; t < 7; ++t) f1[t] = h1o[6 - t];

    const int hrow  = lane & 15;
    const int khalf = (lane >> 4) << 1;
#pragma unroll
    for (int c = 0; c < 6; ++c) {
        int j0 = c * 4 + khalf, j1 = j0 + 1;
        float* o = tA + (size_t)(c * 32 + lane) * 4;
        // T0[r,j] = f0[j-r-1] (h0o: halo-m2 = 1), T1[r,j] = f1[j-r] (h1o: 0)
        o[0] = tapsel<5>(f0, j0 - hrow - 1);
        o[1] = tapsel<5>(f0, j1 - hrow - 1);
        o[2] = tapsel<7>(f1, j0 - hrow);
        o[3] = tapsel<7>(f1, j1 - hrow);
    }
}

// ---------------------------------------------------------------------------
// Level 1: fused rowfilter(h0o,h1o) + colfilter(h0o,h1o) + q2c.
// One wave32 per 16x16 output tile. Column filters run as Toeplitz matmuls on
// the WMMA pipe: out(16x16) = T(16x24) * rowfiltered(24x16), 6 chunks of K=4.
// ---------------------------------------------------------------------------
__global__ __launch_bounds__(32)
void dtcwt_level1(const float* __restrict__ x,
                  const float* __restrict__ h0o, const float* __restrict__ h1o,
                  const float* __restrict__ tA,
                  float* __restrict__ ll1, float* __restrict__ yh0)
{
    const int tid = threadIdx.x;
    const int tx = blockIdx.x, ty = blockIdx.y, img = blockIdx.z;
    const int r0 = ty * 16, c0 = tx * 16;
    const int L = 256;

    __shared__ float sIn[22][24];   // input tile + halo(3) each side
    __shared__ float sLo[24][16];   // row-filtered (h0o), rows 22..23 zero pad
    __shared__ float sHi[24][16];   // row-filtered (h1o)
    __shared__ float sB[3][16][16]; // lh, hh, hl for q2c

    // reversed taps in registers (correlation with h[::-1])
    float f0[5], f1[7];
#pragma unroll
    for (int t = 0; t < 5; ++t) f0[t] = h0o[4 - t];
#pragma unroll
    for (int t = 0; t < 7; ++t) f1[t] = h1o[6 - t];

    const float* xi = x + (size_t)img * 65536;
    __builtin_prefetch(xi + (size_t)refl(r0 - 3, L) * 256, 0, 0);

    // load 22x22 tile with symmetric reflection (async memory->LDS, no VGPR bounce)
    for (int idx = tid; idx < 22 * 22; idx += 32) {
        int rr = idx / 22, cc = idx - rr * 22;
        async_cp_b32(&xi[refl(r0 - 3 + rr, L) * 256 + refl(c0 - 3 + cc, L)], &sIn[rr][cc]);
    }
    if (tid < 16) { sLo[22][tid] = 0.f; sLo[23][tid] = 0.f; sHi[22][tid] = 0.f; sHi[23][tid] = 0.f; }
    async_join();
    __syncthreads();

    // row filters over 22 rows x 16 cols (keep row halo for the col filter)
    for (int idx = tid; idx < 22 * 16; idx += 32) {
        int rr = idx >> 4, cc = idx & 15;
        float lo = 0.f, hi = 0.f;
#pragma unroll
        for (int t = 0; t < 5; ++t) lo += f0[t] * sIn[rr][cc + 1 + t]; // m2=2, halo=3
#pragma unroll
        for (int t = 0; t < 7; ++t) hi += f1[t] * sIn[rr][cc + t];     // m2=3
        sLo[rr][cc] = lo;
        sHi[rr][cc] = hi;
    }
    __syncthreads();

    // ---- column filters as WMMA Toeplitz matmuls ----
    const int hrow  = tid & 15;          // A row / B,C col
    const int khalf = (tid >> 4) << 1;   // K sub-offset per half-wave
    const float4* tAf = (const float4*)tA;

    v8f aLL = {}, aLH = {}, aHL = {}, aHH = {};
#pragma unroll
    for (int c = 0; c < 6; ++c) {
        const float4 av = tAf[c * 32 + tid];  // precomputed Toeplitz fragments
        int j0 = c * 4 + khalf, j1 = j0 + 1;
        v2f a0, a1, bLo, bHi;
        a0.x = av.x;  a0.y = av.y;
        a1.x = av.z;  a1.y = av.w;
        bLo.x = sLo[j0][hrow];  bLo.y = sLo[j1][hrow];
        bHi.x = sHi[j0][hrow];  bHi.y = sHi[j1][hrow];

        aLL = __builtin_amdgcn_wmma_f32_16x16x4_f32(false, a0, false, bLo, (short)0, aLL, false, false);
        aLH = __builtin_amdgcn_wmma_f32_16x16x4_f32(false, a1, false, bLo, (short)0, aLH, false, false);
        aHL = __builtin_amdgcn_wmma_f32_16x16x4_f32(false, a0, false, bHi, (short)0, aHL, false, false);
        aHH = __builtin_amdgcn_wmma_f32_16x16x4_f32(false, a1, false, bHi, (short)0, aHH, false, false);
    }

    // store LL tile (intermediate for level 2); C layout: VGPR v -> row v (+8 for hi half)
    {
        const int mbase = (tid >> 4) * 8;
        float* dst = ll1 + (size_t)img * 65536;
#pragma unroll
        for (int v = 0; v < 8; ++v)
            dst[(size_t)(r0 + mbase + v) * 256 + (c0 + hrow)] = aLL[v];
    }
    // spill lh/hh/hl to LDS for the cross-lane q2c gather
#pragma unroll
    for (int v = 0; v < 8; ++v) {
        int rr = (tid >> 4) * 8 + v;
        sB[0][rr][hrow] = aLH[v];
        sB[1][rr][hrow] = aHH[v];
        sB[2][rr][hrow] = aHL[v];
    }
    __syncthreads();

    // q2c -> yh0 (img, 6, 128, 128, 2)
    for (int idx = tid; idx < 3 * 64; idx += 32) {
        int s = idx >> 6, rem = idx & 63;
        int i = rem >> 3, j = rem & 7;
        float a = sB[s][2 * i][2 * j] * SQ2;
        float b = sB[s][2 * i][2 * j + 1] * SQ2;
        float c = sB[s][2 * i + 1][2 * j] * SQ2;
        float d = sB[s][2 * i + 1][2 * j + 1] * SQ2;
        int I = (r0 >> 1) + i, J = (c0 >> 1) + j;
        size_t b1 = ((((size_t)img * 6 + s)     * 128 + I) * 128 + J) * 2;
        size_t b2 = ((((size_t)img * 6 + 5 - s) * 128 + I) * 128 + J) * 2;
        yh0[b1 + 0] = a - d;  yh0[b1 + 1] = b + c;
        yh0[b2 + 0] = a + d;  yh0[b2 + 1] = b - c;
    }
}

// ---------------------------------------------------------------------------
// Levels 2..J: fused rowdfilt(h0b/h0a, h1b/h1a) + coldfilt(x4) + q2c.
// Input LL is SxS; outputs are (S/2)x(S/2) subbands, yh is (S/4)^2 complex x6.
// 64 threads per 16x16 output tile; stride-2 10-tap filters on VALU.
// ---------------------------------------------------------------------------
__global__ __launch_bounds__(64)
void dtcwt_leveln(const float* __restrict__ llIn, int S,
                  const float* __restrict__ h0a, const float* __restrict__ h0b,
                  const float* __restrict__ h1a, const float* __restrict__ h1b,
                  float* __restrict__ llOut, float* __restrict__ yhj)
{
    const int tid = threadIdx.x;
    const int img = blockIdx.z;
    const int u0 = blockIdx.y * 16, v0 = blockIdx.x * 16; // tile origin in (S/2) grid
    const int n0 = u0 >> 1, q0 = v0 >> 1;
    const int S2 = S >> 1, Sq = S >> 2;

    __shared__ float sX[48][49];   // input window (48x48 orig samples, +1 pad)
    __shared__ float sLo[48][16];  // row-dfilt lowpass
    __shared__ float sHi[48][16];  // row-dfilt highpass
    __shared__ float sB[3][16][16];

    float g0a[10], g0b[10], g1a[10], g1b[10]; // reversed taps
#pragma unroll
    for (int t = 0; t < 10; ++t) {
        g0a[t] = h0a[9 - t]; g0b[t] = h0b[9 - t];
        g1a[t] = h1a[9 - t]; g1b[t] = h1b[9 - t];
    }

    const float* xi = llIn + (size_t)img * S * S;

    // window: orig rows/cols 4*n0-8+s / 4*q0-8+s, s in [0,48)  (async -> LDS)
    for (int idx = tid; idx < 48 * 48; idx += 64) {
        int rr = idx / 48, cc = idx - rr * 48;
        async_cp_b32(&xi[(size_t)refl(4 * n0 - 8 + rr, S) * S + refl(4 * q0 - 8 + cc, S)],
                     &sX[rr][cc]);
    }
    async_join();
    __syncthreads();

    // row-dfilt: lo/hi on 48 rows x 16 out cols.
    //  lo[2q]   = sum g0b[k]*Xp[3+4q+2k] -> sCol = 4*(vc>>1)+1+2k
    //  lo[2q+1] = sum g0a[k]*Xp[2+4q+2k] -> sCol = 4*(vc>>1)+2k
    //  hi[2q]   = sum g1a[k]*Xp[2+4q+2k];  hi[2q+1] = sum g1b[k]*Xp[3+4q+2k]
    for (int idx = tid; idx < 48 * 16; idx += 64) {
        int rr = idx >> 4, vc = idx & 15;
        int sb = (vc >> 1) << 2;
        float lo = 0.f, hi = 0.f;
        if ((vc & 1) == 0) {
#pragma unroll
            for (int k = 0; k < 10; ++k) {
                lo += g0b[k] * sX[rr][sb + 1 + 2 * k];
                hi += g1a[k] * sX[rr][sb + 2 * k];
            }
        } else {
#pragma unroll
            for (int k = 0; k < 10; ++k) {
                lo += g0a[k] * sX[rr][sb + 2 * k];
                hi += g1b[k] * sX[rr][sb + 1 + 2 * k];
            }
        }
        sLo[rr][vc] = lo;
        sHi[rr][vc] = hi;
    }
    __syncthreads();

    // col-dfilt: w = 0:lh(lo,h1) 1:hh(hi,h1) 2:hl(hi,h0) 3:ll(lo,h0)
    for (int idx = tid; idx < 4 * 256; idx += 64) {
        int w = idx >> 8, rem = idx & 255;
        int uc = rem >> 4, vc = rem & 15;
        int sb = (uc >> 1) << 2;
        int odd = uc & 1;
        float acc = 0.f;
        if (w <= 1) { // h1 pair: even -> g1a @ sb+2k ; odd -> g1b @ sb+1+2k
            const float* src0 = (w == 0) ? &sLo[0][0] : &sHi[0][0];
            if (!odd) {
#pragma unroll
                for (int k = 0; k < 10; ++k) acc += g1a[k] * src0[(sb + 2 * k) * 16 + vc];
            } else {
#pragma unroll
                for (int k = 0; k < 10; ++k) acc += g1b[k] * src0[(sb + 1 + 2 * k) * 16 + vc];
            }
        } else {      // h0 pair: even -> g0b @ sb+1+2k ; odd -> g0a @ sb+2k
            const float* src0 = (w == 2) ? &sHi[0][0] : &sLo[0][0];
            if (!odd) {
#pragma unroll
                for (int k = 0; k < 10; ++k) acc += g0b[k] * src0[(sb + 1 + 2 * k) * 16 + vc];
            } else {
#pragma unroll
                for (int k = 0; k < 10; ++k) acc += g0a[k] * src0[(sb + 2 * k) * 16 + vc];
            }
        }
        if (w == 3) {
            llOut[(size_t)img * S2 * S2 + (size_t)(u0 + uc) * S2 + (v0 + vc)] = acc;
        } else {
            sB[w][uc][vc] = acc;
        }
    }
    __syncthreads();

    // q2c -> yhj (img, 6, S/4, S/4, 2)
    for (int idx = tid; idx < 3 * 64; idx += 64) {
        int s = idx >> 6, rem = idx & 63;
        int i = rem >> 3, j = rem & 7;
        float a = sB[s][2 * i][2 * j] * SQ2;
        float b = sB[s][2 * i][2 * j + 1] * SQ2;
        float c = sB[s][2 * i + 1][2 * j] * SQ2;
        float d = sB[s][2 * i + 1][2 * j + 1] * SQ2;
        int I = (u0 >> 1) + i, J = (v0 >> 1) + j;
        size_t b1 = ((((size_t)img * 6 + s)     * Sq + I) * Sq + J) * 2;
        size_t b2 = ((((size_t)img * 6 + 5 - s) * Sq + I) * Sq + J) * 2;
        yhj[b1 + 0] = a - d;  yhj[b1 + 1] = b + c;
        yhj[b2 + 0] = a + d;  yhj[b2 + 1] = b - c;
    }
}

extern "C" void kernel_launch(void* const* d_in, const int* in_sizes, int n_in,
                              void* d_out, int out_size, void* d_ws, size_t ws_size,
                              hipStream_t stream) {
    (void)in_sizes; (void)n_in; (void)out_size; (void)ws_size;
    const float* x   = (const float*)d_in[0];
    const float* h0o = (const float*)d_in[1];
    const float* h1o = (const float*)d_in[2];
    const float* h0a = (const float*)d_in[3];
    const float* h0b = (const float*)d_in[4];
    const float* h1a = (const float*)d_in[5];
    const float* h1b = (const float*)d_in[6];

    float* out    = (float*)d_out;
    float* ll_out = out;                 // (8,16,64,64)        = 524288
    float* yh0    = out + 524288;        // (8,16,6,128,128,2)  = 25165824
    float* yh1    = out + 25690112;      // (8,16,6,64,64,2)    = 6291456
    float* yh2    = out + 31981568;      // (8,16,6,32,32,2)    = 1572864

    float* ll1 = (float*)d_ws;           // 128 * 256 * 256
    float* ll2 = ll1 + 8388608;          // 128 * 128 * 128
    float* tA  = ll2 + 2097152;          // 6 * 32 * 4 Toeplitz A-fragments

    dtcwt_prep<<<dim3(1, 1, 1), 32, 0, stream>>>(h0o, h1o, tA);

    dim3 g1(16, 16, 128);
    dtcwt_level1<<<g1, 32, 0, stream>>>(x, h0o, h1o, tA, ll1, yh0);

    dim3 g2(8, 8, 128);   // S=256 -> out 128x128
    dtcwt_leveln<<<g2, 64, 0, stream>>>(ll1, 256, h0a, h0b, h1a, h1b, ll2, yh1);

    dim3 g3(4, 4, 128);   // S=128 -> out 64x64, ll goes to d_out
    dtcwt_leveln<<<g3, 64, 0, stream>>>(ll2, 128, h0a, h0b, h1a, h1b, ll_out, yh2);
}